// GAT_73976516706474
// MI455X (gfx1250) — compile-verified
//
#include <hip/hip_runtime.h>
#include <cstdint>
#include <cstddef>

// ---------------------------------------------------------------------------
// GATv2 x2 + mean-pool + linear for MI455X (gfx1250, wave32, WMMA)
// ---------------------------------------------------------------------------

typedef __attribute__((ext_vector_type(2))) float v2f;
typedef __attribute__((ext_vector_type(4))) float v4f;
typedef __attribute__((ext_vector_type(8))) float v8f;

template <int H> struct VecT;
template <> struct VecT<128> { using T = v4f; };   // 4 floats/lane -> global_load_b128
template <> struct VecT<64>  { using T = v2f; };   // 2 floats/lane -> global_load_b64

__device__ __forceinline__ float waveReduceSum(float v) {
  #pragma unroll
  for (int m = 16; m >= 1; m >>= 1) v += __shfl_xor(v, m, 32);
  return v;
}
__device__ __forceinline__ float waveReduceMax(float v) {
  #pragma unroll
  for (int m = 16; m >= 1; m >>= 1) v = fmaxf(v, __shfl_xor(v, m, 32));
  return v;
}

// ---------------------------------------------------------------------------
// Y[N,H] = X[N,K] @ W[K,H] + bias[H]   (fp32 WMMA 16x16x4; N%16==0, K%4==0,
// H%16==0). One wave computes one 16x16 tile of Y.
// A 16x4 layout: lanes 0-15 = M rows; VGPR0 = K0|K2 halves, VGPR1 = K1|K3.
// B 4x16 layout: lanes = N cols; VGPR0 = rows K0|K2, VGPR1 = rows K1|K3.
// C/D: VGPR j -> M=j (lanes 0-15) / M=j+8 (lanes 16-31), lane%16 = N.
// ---------------------------------------------------------------------------
__global__ void gemm_bias_k(const float* __restrict__ X,
                            const float* __restrict__ W,
                            const float* __restrict__ bias,
                            float* __restrict__ Y,
                            int Nrows, int K, int H) {
  const int wavesPerBlock = blockDim.x >> 5;
  const int waveId = blockIdx.x * wavesPerBlock + (threadIdx.x >> 5);
  const int lane = threadIdx.x & 31;
  const int colTiles = H >> 4;
  const int rowTile = waveId / colTiles;
  const int colTile = waveId % colTiles;
  if (rowTile * 16 >= Nrows) return;          // wave-uniform exit (EXEC all-1 in body)

  const int row0 = rowTile << 4;
  const int col0 = colTile << 4;
  const int l16 = lane & 15;
  const int hiHalf = lane >> 4;               // 0: K {0,1}, 1: K {2,3}

  v8f acc = {};
  for (int k0 = 0; k0 < K; k0 += 4) {
    const int ka = k0 + (hiHalf << 1);
    const float* ap = X + (size_t)(row0 + l16) * K + ka;
    v2f a; a.x = ap[0]; a.y = ap[1];
    const float* bp = W + (size_t)ka * H + col0 + l16;
    v2f b; b.x = bp[0]; b.y = bp[H];
    acc = __builtin_amdgcn_wmma_f32_16x16x4_f32(
        /*neg_a=*/false, a, /*neg_b=*/false, b,
        /*c_mod=*/(short)0, acc, /*reuse_a=*/false, /*reuse_b=*/false);
  }

  const int m0 = hiHalf << 3;
  const float bval = bias[col0 + l16];
  #pragma unroll
  for (int j = 0; j < 8; ++j) {
    Y[(size_t)(row0 + m0 + j) * H + col0 + l16] = acc[j] + bval;
  }
}

// ---------------------------------------------------------------------------
// Degree + incoming edge-attr sum (for self-loop fill_value='mean')
// ---------------------------------------------------------------------------
__global__ void deg_k(const long long* __restrict__ dst,
                      const float* __restrict__ eattr,
                      int* __restrict__ deg, float* __restrict__ asum, int E_) {
  int e = blockIdx.x * blockDim.x + threadIdx.x;
  if (e >= E_) return;
  int d = (int)dst[e];
  atomicAdd(&deg[d], 1);
  atomicAdd(&asum[d], eattr[e]);
}

__global__ void loopattr_k(const int* __restrict__ deg,
                           const float* __restrict__ asum,
                           float* __restrict__ loop_attr, int Nn) {
  int i = blockIdx.x * blockDim.x + threadIdx.x;
  if (i >= Nn) return;
  loop_attr[i] = asum[i] / fmaxf((float)deg[i], 1.0f);
}

// ---------------------------------------------------------------------------
// Exclusive prefix sum of deg -> rowoff[0..N]  (single block, chunked scan)
// ---------------------------------------------------------------------------
__global__ void scan_k(const int* __restrict__ deg, int* __restrict__ rowoff, int n) {
  __shared__ int sdata[1024];
  __shared__ int carry;
  const int tid = threadIdx.x;
  if (tid == 0) carry = 0;
  __syncthreads();
  for (int base = 0; base < n; base += 1024) {
    int i = base + tid;
    sdata[tid] = (i < n) ? deg[i] : 0;
    __syncthreads();
    for (int off = 1; off < 1024; off <<= 1) {
      int t = (tid >= off) ? sdata[tid - off] : 0;
      __syncthreads();
      sdata[tid] += t;
      __syncthreads();
    }
    if (i < n) rowoff[i + 1] = carry + sdata[tid];
    __syncthreads();
    if (tid == 0) carry += sdata[1023];
    __syncthreads();
  }
  if (tid == 0) rowoff[0] = 0;
}

// ---------------------------------------------------------------------------
// CSR fill (incoming edges grouped by dst)
// ---------------------------------------------------------------------------
__global__ void csrfill_k(const long long* __restrict__ src,
                          const long long* __restrict__ dst,
                          int* __restrict__ fill, const int* __restrict__ rowoff,
                          int* __restrict__ csr_src, int* __restrict__ csr_eid, int E_) {
  int e = blockIdx.x * blockDim.x + threadIdx.x;
  if (e >= E_) return;
  int d = (int)dst[e];
  int pos = rowoff[d] + atomicAdd(&fill[d], 1);
  csr_src[pos] = (int)src[e];
  csr_eid[pos] = e;
}

// ---------------------------------------------------------------------------
// Per-edge attention logits: leaky_relu(xl[src]+xr[dst]+ea*We, 0.2) . att
// One wave per edge; feature dim striped as lane*F..lane*F+F-1 so each lane
// gathers its slice with ONE b128 (H=128) / b64 (H=64) load.
// ---------------------------------------------------------------------------
template <int H>
__global__ void edge_logits_k(const long long* __restrict__ src,
                              const long long* __restrict__ dst,
                              const float* __restrict__ eattr,
                              const float* __restrict__ XL,
                              const float* __restrict__ XR,
                              const float* __restrict__ We,
                              const float* __restrict__ att,
                              float* __restrict__ logits, int E_) {
  int wid = (blockIdx.x * blockDim.x + threadIdx.x) >> 5;
  if (wid >= E_) return;
  const int lane = threadIdx.x & 31;
  constexpr int F = H / 32;
  using vec = typename VecT<H>::T;
  const long long s = src[wid];
  const long long d = dst[wid];
  const float ea = eattr[wid];

  const vec xl = *(const vec*)(XL + (size_t)s * H + lane * F);
  const vec xr = *(const vec*)(XR + (size_t)d * H + lane * F);
  const vec we = *(const vec*)(We + lane * F);
  const vec at = *(const vec*)(att + lane * F);

  float p = 0.f;
  #pragma unroll
  for (int j = 0; j < F; ++j) {
    float m = xl[j] + xr[j] + ea * we[j];
    m = (m > 0.f) ? m : 0.2f * m;
    p += m * at[j];
  }
  p = waveReduceSum(p);
  if (lane == 0) logits[wid] = p;
}

// Self-loop logits (src == dst == node, attr = loop_attr[node])
template <int H>
__global__ void loop_logits_k(const float* __restrict__ loop_attr,
                              const float* __restrict__ XL,
                              const float* __restrict__ XR,
                              const float* __restrict__ We,
                              const float* __restrict__ att,
                              float* __restrict__ loop_lg, int Nn) {
  int wid = (blockIdx.x * blockDim.x + threadIdx.x) >> 5;
  if (wid >= Nn) return;
  const int lane = threadIdx.x & 31;
  constexpr int F = H / 32;
  using vec = typename VecT<H>::T;
  const float ea = loop_attr[wid];

  const vec xl = *(const vec*)(XL + (size_t)wid * H + lane * F);
  const vec xr = *(const vec*)(XR + (size_t)wid * H + lane * F);
  const vec we = *(const vec*)(We + lane * F);
  const vec at = *(const vec*)(att + lane * F);

  float p = 0.f;
  #pragma unroll
  for (int j = 0; j < F; ++j) {
    float m = xl[j] + xr[j] + ea * we[j];
    m = (m > 0.f) ? m : 0.2f * m;
    p += m * at[j];
  }
  p = waveReduceSum(p);
  if (lane == 0) loop_lg[wid] = p;
}

// ---------------------------------------------------------------------------
// Per-node softmax + weighted aggregation + bias + ELU.  One wave per node:
//   pass 1: segment max over {edges into node} U {self loop}
//   pass 2: denom = sum exp, acc = sum exp * xl[src]; out = acc/denom + b; ELU
// Each lane owns a contiguous F-float slice -> one b128/b64 gather per edge.
// ---------------------------------------------------------------------------
template <int H>
__global__ void node_agg_k(const int* __restrict__ rowoff,
                           const int* __restrict__ csr_src,
                           const int* __restrict__ csr_eid,
                           const float* __restrict__ logits,
                           const float* __restrict__ loop_lg,
                           const float* __restrict__ XL,
                           const float* __restrict__ bias,
                           float* __restrict__ OUT, int Nn) {
  int wid = (blockIdx.x * blockDim.x + threadIdx.x) >> 5;
  if (wid >= Nn) return;
  const int lane = threadIdx.x & 31;
  constexpr int F = H / 32;
  using vec = typename VecT<H>::T;
  const int beg = rowoff[wid];
  const int end = rowoff[wid + 1];

  // pass 1: max (lane-parallel over edge list, then wave reduce)
  float mx = loop_lg[wid];
  for (int p = beg + lane; p < end; p += 32) mx = fmaxf(mx, logits[csr_eid[p]]);
  mx = waveReduceMax(mx);

  const float el = __expf(loop_lg[wid] - mx);
  const vec self = *(const vec*)(XL + (size_t)wid * H + lane * F);
  vec acc;
  #pragma unroll
  for (int j = 0; j < F; ++j) acc[j] = el * self[j];
  float denom = el;

  // pass 2: all lanes walk edges together (each lane owns its feature slice)
  for (int p = beg; p < end; ++p) {
    const int e = csr_eid[p];
    const float w = __expf(logits[e] - mx);
    denom += w;
    const int s = csr_src[p];
    const vec xs = *(const vec*)(XL + (size_t)s * H + lane * F);
    #pragma unroll
    for (int j = 0; j < F; ++j) acc[j] += w * xs[j];
  }

  const float inv = 1.f / denom;
  const vec bv = *(const vec*)(bias + lane * F);
  vec outv;
  #pragma unroll
  for (int j = 0; j < F; ++j) {
    float v = acc[j] * inv + bv[j];
    outv[j] = (v > 0.f) ? v : (__expf(v) - 1.f);   // ELU
  }
  *(vec*)(OUT + (size_t)wid * H + lane * F) = outv;
}

// ---------------------------------------------------------------------------
// Mean pool per graph (batch is sorted int64) then linear H2 -> 1
// ---------------------------------------------------------------------------
__global__ void pool_k(const long long* __restrict__ batch,
                       const float* __restrict__ Hfeat,
                       float* __restrict__ pooled, int* __restrict__ cnt, int Nn) {
  int t = blockIdx.x * blockDim.x + threadIdx.x;
  if (t >= Nn * 64) return;
  int node = t >> 6;
  int h = t & 63;
  int g = (int)batch[node];
  atomicAdd(&pooled[g * 64 + h], Hfeat[(size_t)node * 64 + h]);
  if (h == 0) atomicAdd(&cnt[g], 1);
}

__global__ void final_k(const float* __restrict__ pooled,
                        const int* __restrict__ cnt,
                        const float* __restrict__ W3,
                        const float* __restrict__ b3,
                        float* __restrict__ out, int G_) {
  int wid = (blockIdx.x * blockDim.x + threadIdx.x) >> 5;
  if (wid >= G_) return;
  const int lane = threadIdx.x & 31;
  const float inv = 1.f / fmaxf((float)cnt[wid], 1.0f);
  const v2f pv = *(const v2f*)(pooled + wid * 64 + lane * 2);
  const v2f wv = *(const v2f*)(W3 + lane * 2);
  float p = pv.x * wv.x + pv.y * wv.y;
  p = waveReduceSum(p);
  if (lane == 0) out[wid] = p * inv + b3[0];
}

// ---------------------------------------------------------------------------
// Host launch
// ---------------------------------------------------------------------------
extern "C" void kernel_launch(void* const* d_in, const int* in_sizes, int n_in,
                              void* d_out, int out_size, void* d_ws, size_t ws_size,
                              hipStream_t stream) {
  const float* x           = (const float*)d_in[0];
  const long long* eidx    = (const long long*)d_in[1];
  const float* edge_attr   = (const float*)d_in[2];
  const long long* batch   = (const long long*)d_in[3];
  const float* Wl1 = (const float*)d_in[4];  const float* bl1 = (const float*)d_in[5];
  const float* Wr1 = (const float*)d_in[6];  const float* br1 = (const float*)d_in[7];
  const float* We1 = (const float*)d_in[8];  const float* att1= (const float*)d_in[9];
  const float* b1  = (const float*)d_in[10];
  const float* Wl2 = (const float*)d_in[11]; const float* bl2 = (const float*)d_in[12];
  const float* Wr2 = (const float*)d_in[13]; const float* br2 = (const float*)d_in[14];
  const float* We2 = (const float*)d_in[15]; const float* att2= (const float*)d_in[16];
  const float* b2  = (const float*)d_in[17];
  const float* W3  = (const float*)d_in[18]; const float* b3  = (const float*)d_in[19];
  float* out = (float*)d_out;

  const int IN_ = 128, H1_ = 128, H2_ = 64;
  const int N_ = in_sizes[0] / IN_;   // 40000
  const int E_ = in_sizes[1] / 2;     // 640000
  const int G_ = out_size;            // 64

  // ---- workspace carve (regions re-used across layers; ~70 MB) ----
  char* w = (char*)d_ws;
  size_t off = 0;
  auto carve = [&](size_t bytes) -> char* {
    char* p = w + off;
    off += (bytes + 255) & ~(size_t)255;
    return p;
  };
  int*   deg       = (int*)  carve((size_t)N_ * 4);
  float* asum      = (float*)carve((size_t)N_ * 4);
  int*   fill      = (int*)  carve((size_t)N_ * 4);
  float* pooled    = (float*)carve((size_t)G_ * H2_ * 4);
  int*   cnt       = (int*)  carve((size_t)G_ * 4);
  const size_t zeroBytes = off;                    // everything above needs zeroing
  int*   rowoff    = (int*)  carve((size_t)(N_ + 1) * 4);
  float* loop_attr = (float*)carve((size_t)N_ * 4);
  int*   csr_src   = (int*)  carve((size_t)E_ * 4);
  int*   csr_eid   = (int*)  carve((size_t)E_ * 4);
  float* logits    = (float*)carve((size_t)E_ * 4);
  float* loop_lg   = (float*)carve((size_t)N_ * 4);
  float* regA      = (float*)carve((size_t)N_ * H1_ * 4);   // XL1, later XL2 + H2out
  float* regB      = (float*)carve((size_t)N_ * H1_ * 4);   // XR1, later XR2
  float* Hbuf      = (float*)carve((size_t)N_ * H1_ * 4);   // layer-1 output
  float* XL2   = regA;
  float* H2out = regA + (size_t)N_ * H2_;
  float* XR2   = regB;
  (void)ws_size; (void)n_in;

  hipMemsetAsync(d_ws, 0, zeroBytes, stream);

  const long long* srcp = eidx;
  const long long* dstp = eidx + E_;

  // graph preprocessing
  deg_k     <<<(E_ + 255) / 256, 256, 0, stream>>>(dstp, edge_attr, deg, asum, E_);
  loopattr_k<<<(N_ + 255) / 256, 256, 0, stream>>>(deg, asum, loop_attr, N_);
  scan_k    <<<1, 1024, 0, stream>>>(deg, rowoff, N_);
  csrfill_k <<<(E_ + 255) / 256, 256, 0, stream>>>(srcp, dstp, fill, rowoff, csr_src, csr_eid, E_);

  // ---- layer 1: GATv2 (IN=128 -> H1=128) ----
  {
    const int waves = ((N_ + 15) / 16) * (H1_ / 16);
    gemm_bias_k<<<(waves + 7) / 8, 256, 0, stream>>>(x, Wl1, bl1, regA, N_, IN_, H1_);
    gemm_bias_k<<<(waves + 7) / 8, 256, 0, stream>>>(x, Wr1, br1, regB, N_, IN_, H1_);
  }
  edge_logits_k<128><<<(E_ + 7) / 8, 256, 0, stream>>>(srcp, dstp, edge_attr, regA, regB, We1, att1, logits, E_);
  loop_logits_k<128><<<(N_ + 7) / 8, 256, 0, stream>>>(loop_attr, regA, regB, We1, att1, loop_lg, N_);
  node_agg_k  <128><<<(N_ + 7) / 8, 256, 0, stream>>>(rowoff, csr_src, csr_eid, logits, loop_lg, regA, b1, Hbuf, N_);

  // ---- layer 2: GATv2 (H1=128 -> H2=64) ----
  {
    const int waves = ((N_ + 15) / 16) * (H2_ / 16);
    gemm_bias_k<<<(waves + 7) / 8, 256, 0, stream>>>(Hbuf, Wl2, bl2, XL2, N_, H1_, H2_);
    gemm_bias_k<<<(waves + 7) / 8, 256, 0, stream>>>(Hbuf, Wr2, br2, XR2, N_, H1_, H2_);
  }
  edge_logits_k<64><<<(E_ + 7) / 8, 256, 0, stream>>>(srcp, dstp, edge_attr, XL2, XR2, We2, att2, logits, E_);
  loop_logits_k<64><<<(N_ + 7) / 8, 256, 0, stream>>>(loop_attr, XL2, XR2, We2, att2, loop_lg, N_);
  node_agg_k  <64><<<(N_ + 7) / 8, 256, 0, stream>>>(rowoff, csr_src, csr_eid, logits, loop_lg, XL2, b2, H2out, N_);

  // ---- mean pool + linear ----
  pool_k <<<((size_t)N_ * 64 + 255) / 256, 256, 0, stream>>>(batch, H2out, pooled, cnt, N_);
  final_k<<<(G_ + 7) / 8, 256, 0, stream>>>(pooled, cnt, W3, b3, out, G_);
}